// PointNetFeaturePropagation_85358180040864
// MI455X (gfx1250) — compile-verified
//
#include <hip/hip_runtime.h>
#include <hip/hip_bf16.h>
#include <math.h>

// ---------------------------------------------------------------------------
// PointNetFeaturePropagation for MI455X (gfx1250, wave32, WMMA f32).
// Pipeline:
//   pack_w (x2)  -> WMMA-fragment-ordered weight repack (coalesced A loads)
//   knn_interp_concat -> x1[b,384,n]   (float4 LDS point cache, 1 ds_load/iter)
//   gemm_wmma<384,false> -> y1
//   bn_stats -> scale1/shift1
//   gemm_wmma<256,true>  -> y2   (BN1 apply + ReLU fused into B-fragment load)
//   bn_stats -> scale2/shift2
//   bn_relu  -> d_out (non-temporal b128 stores)
// ---------------------------------------------------------------------------

typedef __attribute__((ext_vector_type(2))) float v2f;
typedef __attribute__((ext_vector_type(4))) float v4f;
typedef __attribute__((ext_vector_type(8))) float v8f;

namespace {
constexpr int kB   = 8;
constexpr int kN   = 8192;   // query points
constexpr int kS   = 2048;   // source points
constexpr int kD1  = 128;    // points1 channels
constexpr int kD2  = 256;    // points2 channels
constexpr int kCin = kD1 + kD2;  // 384
constexpr int kCm  = 256;    // mid/out channels
constexpr float kEps = 1e-5f;
}

// ---------------------------------------------------------------------------
// Repack W[m,k] (row-major, stride K) into WMMA A-fragment order:
//   Wp[((mt*(K/2) + p)*16 + r)*2 + j] = W[(mt*16 + r)*K + 2p + j]
// so a wave's A-fragment load for k-step k0 is one coalesced b64 per lane:
//   lane(l): pair index p = k0/2 + (l>=16), row r = l%16.
// ---------------------------------------------------------------------------
__global__ void pack_w(const float* __restrict__ W, float* __restrict__ Wp, int K) {
  const int idx = blockIdx.x * blockDim.x + threadIdx.x;
  const int total = kCm * K;
  if (idx >= total) return;
  const int K2 = K >> 1;
  const int j = idx & 1;
  const int r = (idx >> 1) & 15;
  const int rest = idx >> 5;        // mt*K2 + p
  const int p = rest % K2;
  const int mt = rest / K2;
  Wp[idx] = W[(size_t)(mt * 16 + r) * K + 2 * p + j];
}

// ---------------------------------------------------------------------------
// Kernel: 3-NN inverse-distance interpolation + concat.
// Block: 256 threads = 256 consecutive query points of one batch.
// xyz2[b] staged in LDS as padded float4 (32 KB) -> one ds_load_b128 per
// candidate point in the hot scan loop (broadcast reads, conflict-free).
// ---------------------------------------------------------------------------
__global__ void knn_interp_concat(const float* __restrict__ xyz1,
                                  const float* __restrict__ xyz2,
                                  const float* __restrict__ points1,
                                  const float* __restrict__ points2,
                                  float* __restrict__ x1) {
  __shared__ v4f sp[kS];

  const int b = blockIdx.y;
  const int n = blockIdx.x * blockDim.x + threadIdx.x;

  const float* p2b = xyz2 + (size_t)b * 3 * kS;
  for (int s = threadIdx.x; s < kS; s += blockDim.x) {
    v4f q;
    q.x = p2b[0 * kS + s];
    q.y = p2b[1 * kS + s];
    q.z = p2b[2 * kS + s];
    q.w = 0.0f;
    sp[s] = q;
  }
  __syncthreads();

  const float* p1b = xyz1 + (size_t)b * 3 * kN;
  const float px = p1b[0 * kN + n];
  const float py = p1b[1 * kN + n];
  const float pz = p1b[2 * kN + n];

  float d0 = 3.4e38f, d1 = 3.4e38f, d2 = 3.4e38f;
  int   i0 = 0, i1 = 0, i2 = 0;
  for (int s = 0; s < kS; ++s) {
    const v4f q = sp[s];
    const float dx = px - q.x;
    const float dy = py - q.y;
    const float dz = pz - q.z;
    const float d = dx * dx + dy * dy + dz * dz;
    if (d < d2) {
      if (d < d1) {
        if (d < d0) { d2 = d1; i2 = i1; d1 = d0; i1 = i0; d0 = d; i0 = s; }
        else        { d2 = d1; i2 = i1; d1 = d;  i1 = s; }
      } else        { d2 = d;  i2 = s; }
    }
  }

  const float r0 = 1.0f / (d0 + 1e-8f);
  const float r1 = 1.0f / (d1 + 1e-8f);
  const float r2 = 1.0f / (d2 + 1e-8f);
  const float rs = 1.0f / (r0 + r1 + r2);
  const float w0 = r0 * rs, w1 = r1 * rs, w2 = r2 * rs;

  const float* pt1 = points1 + (size_t)b * kD1 * kN;
  float* xo = x1 + (size_t)b * kCin * kN;
  for (int c = 0; c < kD1; ++c) {
    xo[(size_t)c * kN + n] = pt1[(size_t)c * kN + n];
  }
  const float* pt2 = points2 + (size_t)b * kD2 * kS;
  for (int c = 0; c < kD2; ++c) {
    const float* row = pt2 + (size_t)c * kS;
    xo[(size_t)(kD1 + c) * kN + n] = w0 * row[i0] + w1 * row[i1] + w2 * row[i2];
  }
}

// ---------------------------------------------------------------------------
// GEMM via V_WMMA_F32_16X16X4_F32.
//   Y[b,m,n] = sum_k W[m,k] * Xeff[b,k,n] + bias[m]
//   FUSE: Xeff = relu(X*scale[k] + shift[k])  (BN1 apply fused into load)
// Block: 256 threads = 8 waves. Wave w owns M-tiles 2w, 2w+1 (32 channels),
// block covers a 64-wide n tile -> 2x4 accumulators per wave.
// Per K-step: 2 coalesced A-loads (packed W) + 4 B-loads -> 8 WMMAs.
// A-frag (ISA 16x4 f32): lanes 0-15 K=k0,k0+1; lanes 16-31 K=k0+2,k0+3.
// B-frag mirror: v0 = row K (lower half) / K+2 (upper half), v1 = +1 row.
// C/D: VGPR r -> M = r (+8 upper half), N = lane%16.
// ---------------------------------------------------------------------------
template <int K, bool FUSE>
__global__ void gemm_wmma(const float* __restrict__ Wp,
                          const float* __restrict__ X,
                          const float* __restrict__ bias,
                          const float* __restrict__ scale,
                          const float* __restrict__ shift,
                          float* __restrict__ Y) {
  constexpr int K2 = K / 2;
  __shared__ float ssc[K];
  __shared__ float ssh[K];

  const int b    = blockIdx.y;
  const int n0   = blockIdx.x * 64;
  const int wave = threadIdx.x >> 5;     // 0..7
  const int lane = threadIdx.x & 31;
  const int half = lane >> 4;            // 0: K+0/1   1: K+2/3
  const int lr   = lane & 15;

  if (FUSE) {
    for (int c = threadIdx.x; c < K; c += blockDim.x) {
      ssc[c] = scale[c];
      ssh[c] = shift[c];
    }
    __syncthreads();
  }

  const v2f* wp = (const v2f*)Wp;
  const int mt0 = wave * 2;
  const int mt1 = wave * 2 + 1;
  const float* Xb = X + (size_t)b * K * kN + n0 + lr;

  v8f acc[2][4] = {};

  for (int k0 = 0; k0 < K; k0 += 4) {
    const int p = (k0 >> 1) + half;      // packed k-pair index for this half
    const v2f a0 = wp[((size_t)mt0 * K2 + p) * 16 + lr];
    const v2f a1 = wp[((size_t)mt1 * K2 + p) * 16 + lr];

    const int ka = k0 + half * 2;
    const float* xr0 = Xb + (size_t)ka * kN;
    const float* xr1 = xr0 + kN;
    v2f bf[4];
#pragma unroll
    for (int t = 0; t < 4; ++t) {
      bf[t].x = xr0[t * 16];
      bf[t].y = xr1[t * 16];
    }
    if (FUSE) {
      const float sc0 = ssc[ka],     sh0 = ssh[ka];
      const float sc1 = ssc[ka + 1], sh1 = ssh[ka + 1];
#pragma unroll
      for (int t = 0; t < 4; ++t) {
        bf[t].x = fmaxf(bf[t].x * sc0 + sh0, 0.0f);
        bf[t].y = fmaxf(bf[t].y * sc1 + sh1, 0.0f);
      }
    }
#pragma unroll
    for (int t = 0; t < 4; ++t) {
      acc[0][t] = __builtin_amdgcn_wmma_f32_16x16x4_f32(false, a0, false, bf[t],
                                                        (short)0, acc[0][t], false, false);
      acc[1][t] = __builtin_amdgcn_wmma_f32_16x16x4_f32(false, a1, false, bf[t],
                                                        (short)0, acc[1][t], false, false);
    }
  }

  float* Yb = Y + (size_t)b * kCm * kN + n0 + lr;
#pragma unroll
  for (int mt = 0; mt < 2; ++mt) {
    const int mbase = (wave * 2 + mt) * 16;
#pragma unroll
    for (int r = 0; r < 8; ++r) {
      const int m = mbase + r + half * 8;
      const float bv = bias[m];
      float* yr = Yb + (size_t)m * kN;
      yr[0]  = acc[mt][0][r] + bv;
      yr[16] = acc[mt][1][r] + bv;
      yr[32] = acc[mt][2][r] + bv;
      yr[48] = acc[mt][3][r] + bv;
    }
  }
}

// ---------------------------------------------------------------------------
// Per-channel training-mode BN statistics over (B, N) = 65536 values.
// One block (256 threads) per channel; b128 loads; fused scale/shift out.
// ---------------------------------------------------------------------------
__global__ void bn_stats(const float* __restrict__ Y,
                         const float* __restrict__ g,
                         const float* __restrict__ beta,
                         float* __restrict__ scale,
                         float* __restrict__ shift) {
  const int c = blockIdx.x;
  float s = 0.0f, ss = 0.0f;
  for (int b = 0; b < kB; ++b) {
    const v4f* row = (const v4f*)(Y + ((size_t)b * kCm + c) * kN);
    for (int n = threadIdx.x; n < kN / 4; n += blockDim.x) {
      const v4f v = row[n];
      s  += v.x + v.y + v.z + v.w;
      ss += v.x * v.x + v.y * v.y + v.z * v.z + v.w * v.w;
    }
  }
  __shared__ float rs[256];
  __shared__ float rss[256];
  rs[threadIdx.x] = s;
  rss[threadIdx.x] = ss;
  __syncthreads();
  for (int off = 128; off > 0; off >>= 1) {
    if ((int)threadIdx.x < off) {
      rs[threadIdx.x]  += rs[threadIdx.x + off];
      rss[threadIdx.x] += rss[threadIdx.x + off];
    }
    __syncthreads();
  }
  if (threadIdx.x == 0) {
    const float inv = 1.0f / (float)(kB * kN);
    const float mean = rs[0] * inv;
    const float var  = rss[0] * inv - mean * mean;   // biased, matches torch BN norm
    const float rstd = rsqrtf(var + kEps);
    const float sc = g[c] * rstd;
    scale[c] = sc;
    shift[c] = beta[c] - mean * sc;
  }
}

// ---------------------------------------------------------------------------
// Elementwise BN apply + ReLU, b128 in / b128 non-temporal out.
// All 4 elements of a v4f share one channel (kN % 4 == 0).
// ---------------------------------------------------------------------------
__global__ void bn_relu(const float* __restrict__ Y,
                        const float* __restrict__ scale,
                        const float* __restrict__ shift,
                        float* __restrict__ Out) {
  const size_t i4 = (size_t)blockIdx.x * blockDim.x + threadIdx.x;
  const int c = (int)((i4 * 4 / kN) % kCm);
  const float sc = scale[c];
  const float sh = shift[c];
  const v4f v = ((const v4f*)Y)[i4];
  v4f o;
  o.x = fmaxf(v.x * sc + sh, 0.0f);
  o.y = fmaxf(v.y * sc + sh, 0.0f);
  o.z = fmaxf(v.z * sc + sh, 0.0f);
  o.w = fmaxf(v.w * sc + sh, 0.0f);
  __builtin_nontemporal_store(o, (v4f*)Out + i4);
}

// ---------------------------------------------------------------------------
extern "C" void kernel_launch(void* const* d_in, const int* in_sizes, int n_in,
                              void* d_out, int out_size, void* d_ws, size_t ws_size,
                              hipStream_t stream) {
  (void)in_sizes; (void)n_in; (void)out_size; (void)ws_size;

  const float* xyz1    = (const float*)d_in[0];
  const float* xyz2    = (const float*)d_in[1];
  const float* points1 = (const float*)d_in[2];
  const float* points2 = (const float*)d_in[3];
  const float* w1      = (const float*)d_in[4];
  const float* b1      = (const float*)d_in[5];
  const float* g1      = (const float*)d_in[6];
  const float* beta1   = (const float*)d_in[7];
  const float* w2      = (const float*)d_in[8];
  const float* b2      = (const float*)d_in[9];
  const float* g2      = (const float*)d_in[10];
  const float* beta2   = (const float*)d_in[11];

  // Workspace layout (floats). y2 reuses the x1 region (dead after GEMM1).
  float* ws = (float*)d_ws;
  const size_t x1_elems = (size_t)kB * kCin * kN;  // 25.2 M (100.7 MB)
  const size_t yz_elems = (size_t)kB * kCm * kN;   // 16.8 M ( 67.1 MB)
  float* x1 = ws;
  float* y1 = x1 + x1_elems;
  float* stats = y1 + yz_elems;                    // 4 x 256 floats
  float* wp1 = stats + 1024;                       // 256*384 packed weights
  float* wp2 = wp1 + (size_t)kCm * kCin;           // 256*256 packed weights
  float* y2 = ws;                                  // reuse x1 region
  float* scale1 = stats;
  float* shift1 = stats + 256;
  float* scale2 = stats + 512;
  float* shift2 = stats + 768;

  // 0) Repack weights into WMMA A-fragment order (coalesced A loads).
  pack_w<<<(kCm * kCin + 255) / 256, 256, 0, stream>>>(w1, wp1, kCin);
  pack_w<<<(kCm * kCm + 255) / 256, 256, 0, stream>>>(w2, wp2, kCm);

  // 1) kNN interpolation + concat -> x1[b, 384, n]
  {
    dim3 grid(kN / 256, kB);
    knn_interp_concat<<<grid, 256, 0, stream>>>(xyz1, xyz2, points1, points2, x1);
  }
  // 2) GEMM1 (K = 384) -> y1
  {
    dim3 grid(kN / 64, kB);
    gemm_wmma<kCin, false><<<grid, 256, 0, stream>>>(wp1, x1, b1, scale1, shift1, y1);
  }
  // 3) BN1 stats -> scale1/shift1
  bn_stats<<<kCm, 256, 0, stream>>>(y1, g1, beta1, scale1, shift1);
  // 4) GEMM2 (K = 256), BN1 apply + ReLU fused into B-fragment loads -> y2
  {
    dim3 grid(kN / 64, kB);
    gemm_wmma<kCm, true><<<grid, 256, 0, stream>>>(wp2, y1, b2, scale1, shift1, y2);
  }
  // 5) BN2 stats -> scale2/shift2
  bn_stats<<<kCm, 256, 0, stream>>>(y2, g2, beta2, scale2, shift2);
  // 6) BN2 apply + ReLU -> d_out (non-temporal)
  bn_relu<<<(unsigned)(yz_elems / 4 / 256), 256, 0, stream>>>(y2, scale2, shift2,
                                                              (float*)d_out);
}